// TLGv4SelfAttention_7696581394724
// MI455X (gfx1250) — compile-verified
//
#include <hip/hip_runtime.h>
#include <math.h>

typedef float v2f __attribute__((ext_vector_type(2)));
typedef float v8f __attribute__((ext_vector_type(8)));

#define S_LEN 2048
#define HIDN  2048
#define NHQ   32
#define NHKV  8
#define DH    64
#define NQKV  3072        // (32 + 2*8) * 64
#define NBLK  32          // S / 64
#define LOCALW 16
#define VERTP  8
#define ATT_SCALE 0.125f  // 1/sqrt(64)

static __device__ __forceinline__ v8f wmma4(v2f a, v2f b, v8f c) {
    // V_WMMA_F32_16X16X4_F32 : full-precision fp32 matrix op (CDNA5)
    return __builtin_amdgcn_wmma_f32_16x16x4_f32(
        /*neg_a=*/false, a, /*neg_b=*/false, b,
        /*c_mod=*/(short)0, c, /*reuse_a=*/false, /*reuse_b=*/false);
}

// Async global -> LDS copy of 16 bytes (GLOBAL_LOAD_ASYNC_TO_LDS_B128, ASYNCcnt).
// LDS address = low 32 bits of the generic pointer (flat aperture rule).
static __device__ __forceinline__ void async_copy16(void* lds_dst, const void* gsrc) {
    unsigned ldsa = (unsigned)(uintptr_t)lds_dst;
    asm volatile("global_load_async_to_lds_b128 %0, %1, off"
                 :: "v"(ldsa), "v"(gsrc) : "memory");
}
static __device__ __forceinline__ void wait_async0() {
    asm volatile("s_wait_asynccnt 0x0" ::: "memory");
}

// ---------------------------------------------------------------------------
// C[M,N] = A[M,K] * W[N,K]^T + bias[N]     (both A and W are K-major)
// 128 threads = 4 waves; WG tile 64x64; each wave 2x2 subtiles of 16x16.
// Double-buffered LDS tiles filled by async-to-LDS DMA, overlapped with WMMA.
// ---------------------------------------------------------------------------
__global__ __launch_bounds__(128)
void gemm_nt_bias(const float* __restrict__ A, const float* __restrict__ W,
                  const float* __restrict__ bias, float* __restrict__ C,
                  int M, int N, int K) {
    const int KT = 32;
    __shared__ float As[2][64][36];   // pitch 36: 16B-aligned rows, conflict-free frag reads
    __shared__ float Bs[2][64][36];

    const int tileM = blockIdx.y * 64;
    const int tileN = blockIdx.x * 64;
    const int t    = threadIdx.x;
    const int lane = t & 31;
    const int w    = t >> 5;
    const int wm   = w >> 1, wn = w & 1;
    const int n16  = lane & 15;
    const int half = lane >> 4;

    v8f cacc[2][2];
#pragma unroll
    for (int ti = 0; ti < 2; ++ti)
#pragma unroll
        for (int tj = 0; tj < 2; ++tj)
#pragma unroll
            for (int r = 0; r < 8; ++r) cacc[ti][tj][r] = 0.0f;

    auto issue_tile = [&](int buf, int k0) {
#pragma unroll
        for (int rep = 0; rep < 4; ++rep) {
            int idx = rep * 128 + t;
            int row = idx >> 3;            // 8 float4 per 32-wide row
            int c4  = (idx & 7) << 2;
            async_copy16(&As[buf][row][c4], A + (size_t)(tileM + row) * K + k0 + c4);
            async_copy16(&Bs[buf][row][c4], W + (size_t)(tileN + row) * K + k0 + c4);
        }
    };

    const int ntiles = K / KT;
    issue_tile(0, 0);

    for (int p = 0; p < ntiles; ++p) {
        const int cur = p & 1;
        wait_async0();          // this wave's async copies into `cur` complete
        __syncthreads();        // all waves' copies visible
        if (p + 1 < ntiles) issue_tile(cur ^ 1, (p + 1) * KT);  // overlap with compute

#pragma unroll
        for (int ks = 0; ks < KT; ks += 4) {
            int kc = ks + 2 * half;  // K pair per lane (even -> 8B aligned)
            v2f a0 = *(const v2f*)&As[cur][wm * 32 +      n16][kc];
            v2f a1 = *(const v2f*)&As[cur][wm * 32 + 16 + n16][kc];
            v2f b0 = *(const v2f*)&Bs[cur][wn * 32 +      n16][kc];
            v2f b1 = *(const v2f*)&Bs[cur][wn * 32 + 16 + n16][kc];
            cacc[0][0] = wmma4(a0, b0, cacc[0][0]);
            cacc[0][1] = wmma4(a0, b1, cacc[0][1]);
            cacc[1][0] = wmma4(a1, b0, cacc[1][0]);
            cacc[1][1] = wmma4(a1, b1, cacc[1][1]);
        }
        __syncthreads();        // all waves done reading `cur` before it is refilled
    }

#pragma unroll
    for (int ti = 0; ti < 2; ++ti) {
        int rowb = tileM + wm * 32 + ti * 16 + 8 * half;
#pragma unroll
        for (int tj = 0; tj < 2; ++tj) {
            int col = tileN + wn * 32 + tj * 16 + n16;
            float bv = bias[col];
#pragma unroll
            for (int r = 0; r < 8; ++r)
                C[(size_t)(rowb + r) * N + col] = cacc[ti][tj][r] + bv;
        }
    }
}

// ---------------------------------------------------------------------------
// RoPE + head scatter: mixed[S, 8, 6, 64] -> q[32][S][64], k[8][S][64], v[8][S][64]
// ---------------------------------------------------------------------------
__global__ void rope_scatter(const float* __restrict__ mixed,
                             float* __restrict__ qr, float* __restrict__ kr,
                             float* __restrict__ vv) {
    int idx = blockIdx.x * blockDim.x + threadIdx.x;
    if (idx >= S_LEN * NQKV) return;
    int s  = idx / NQKV;
    int c  = idx - s * NQKV;
    int kv = c / (6 * 64);
    int g  = (c / 64) % 6;
    int d  = c & 63;
    float x = mixed[idx];
    if (g == 5) {                       // V: plain copy
        vv[((size_t)kv * S_LEN + s) * 64 + d] = x;
        return;
    }
    int   dd  = d & 31;
    float inv = powf(10000.0f, -((float)(2 * dd)) / 64.0f);
    float ang = (float)s * inv;
    float cs  = cosf(ang);
    float sn  = sinf(ang);
    float other = mixed[(size_t)s * NQKV + (kv * 6 + g) * 64 + ((d < 32) ? d + 32 : d - 32)];
    float outv  = (d < 32) ? (x * cs - other * sn) : (x * cs + other * sn);
    if (g == 4) kr[((size_t)kv * S_LEN + s) * 64 + d] = outv;
    else        qr[(((size_t)(kv * 4 + g)) * S_LEN + s) * 64 + d] = outv;
}

// ---------------------------------------------------------------------------
// Block-sparse flash attention: 1 WG per (q-block i, head h); 4 waves,
// each wave owns 16 query rows. Exact online softmax. All GEMMs via WMMA f32.
// Q and K tiles arrive via async-to-LDS DMA; V is transposed through VGPRs.
// ---------------------------------------------------------------------------
__global__ __launch_bounds__(128)
void attn_kernel(const float* __restrict__ qr, const float* __restrict__ kr,
                 const float* __restrict__ vv, float* __restrict__ ctx) {
    __shared__ float Qs [64][68];   // Q tile, row-major
    __shared__ float KPs[64][68];   // K tile (scores phase) aliased with P tile (PV phase)
    __shared__ float VTs[64][68];   // V tile, transposed: VTs[d][k]

    const int i   = blockIdx.x;     // query block
    const int h   = blockIdx.y;     // q head
    const int kvh = h >> 2;
    const int t    = threadIdx.x;
    const int lane = t & 31;
    const int w    = t >> 5;
    const int n16  = lane & 15;
    const int half = lane >> 4;

    const float* Qg = qr + ((size_t)h * S_LEN + (size_t)i * 64) * 64;
#pragma unroll
    for (int rep = 0; rep < 8; ++rep) {
        int idx = rep * 128 + t;
        int row = idx >> 4;             // 16 float4 per 64-wide row
        int c4  = (idx & 15) << 2;
        async_copy16(&Qs[row][c4], Qg + row * 64 + c4);
    }

    v8f of[4];
    float mrow[8], lrow[8];
#pragma unroll
    for (int nt = 0; nt < 4; ++nt)
#pragma unroll
        for (int r = 0; r < 8; ++r) of[nt][r] = 0.0f;
#pragma unroll
    for (int r = 0; r < 8; ++r) { mrow[r] = -3.0e38f; lrow[r] = 0.0f; }

    for (int j = 0; j <= i; ++j) {
        bool ok = (i - j < LOCALW) || (((j + 1) & (VERTP - 1)) == 0);
        if (!ok) continue;

        __syncthreads();   // previous iteration's PV reads complete
        const float* Kg = kr + ((size_t)kvh * S_LEN + (size_t)j * 64) * 64;
        const float* Vg = vv + ((size_t)kvh * S_LEN + (size_t)j * 64) * 64;
#pragma unroll
        for (int rep = 0; rep < 8; ++rep) {
            int idx = rep * 128 + t;
            int row = idx >> 4;
            int c4  = (idx & 15) << 2;
            async_copy16(&KPs[row][c4], Kg + row * 64 + c4);     // K: async DMA
            float4 v4 = *(const float4*)(Vg + row * 64 + c4);    // V: transpose in VGPRs
            VTs[c4 + 0][row] = v4.x;  VTs[c4 + 1][row] = v4.y;
            VTs[c4 + 2][row] = v4.z;  VTs[c4 + 3][row] = v4.w;
        }
        wait_async0();     // Q (first pass) + K copies for this wave done
        __syncthreads();

        // S = Q K^T  (16x64 per wave; 4 N-subtiles)
        v8f sf[4];
#pragma unroll
        for (int nt = 0; nt < 4; ++nt)
#pragma unroll
            for (int r = 0; r < 8; ++r) sf[nt][r] = 0.0f;
#pragma unroll
        for (int dd = 0; dd < 64; dd += 4) {
            int kc = dd + 2 * half;
            v2f a = *(const v2f*)&Qs[w * 16 + n16][kc];
#pragma unroll
            for (int nt = 0; nt < 4; ++nt) {
                v2f b = *(const v2f*)&KPs[nt * 16 + n16][kc];
                sf[nt] = wmma4(a, b, sf[nt]);
            }
        }

        // scale + causal mask (diag block) + row max
        float mx[8];
#pragma unroll
        for (int r = 0; r < 8; ++r) mx[r] = -3.0e38f;
#pragma unroll
        for (int nt = 0; nt < 4; ++nt)
#pragma unroll
            for (int r = 0; r < 8; ++r) {
                float v = sf[nt][r] * ATT_SCALE;
                if (j == i && (nt * 16 + n16) > (w * 16 + r + 8 * half)) v = -3.0e38f;
                sf[nt][r] = v;
                mx[r] = fmaxf(mx[r], v);
            }
#pragma unroll
        for (int r = 0; r < 8; ++r)
#pragma unroll
            for (int ofs = 1; ofs < 16; ofs <<= 1)   // reduce across the 16-lane half
                mx[r] = fmaxf(mx[r], __shfl_xor(mx[r], ofs, 32));

        float corr[8], rs[8];
#pragma unroll
        for (int r = 0; r < 8; ++r) {
            float mnew = fmaxf(mrow[r], mx[r]);
            corr[r] = __expf(mrow[r] - mnew);
            mrow[r] = mnew;
            rs[r]   = 0.0f;
        }

        __syncthreads();   // all waves done reading K tile before P overwrites it

        // P = exp(S - m), stage via LDS (aliased over K tile), accumulate row sums
#pragma unroll
        for (int nt = 0; nt < 4; ++nt)
#pragma unroll
            for (int r = 0; r < 8; ++r) {
                float p = __expf(sf[nt][r] - mrow[r]);
                rs[r] += p;
                KPs[w * 16 + r + 8 * half][nt * 16 + n16] = p;
            }
#pragma unroll
        for (int r = 0; r < 8; ++r) {
#pragma unroll
            for (int ofs = 1; ofs < 16; ofs <<= 1)
                rs[r] += __shfl_xor(rs[r], ofs, 32);
            lrow[r] = lrow[r] * corr[r] + rs[r];
        }
#pragma unroll
        for (int nt = 0; nt < 4; ++nt)
#pragma unroll
            for (int r = 0; r < 8; ++r) of[nt][r] *= corr[r];

        // O += P V   (B fragments contiguous thanks to transposed V tile)
#pragma unroll
        for (int kk = 0; kk < 64; kk += 4) {
            int kc = kk + 2 * half;
            v2f a = *(const v2f*)&KPs[w * 16 + n16][kc];
#pragma unroll
            for (int nt = 0; nt < 4; ++nt) {
                v2f b = *(const v2f*)&VTs[nt * 16 + n16][kc];
                of[nt] = wmma4(a, b, of[nt]);
            }
        }
    }

    // normalize + write ctx in [S, HID] layout for the dense GEMM
#pragma unroll
    for (int nt = 0; nt < 4; ++nt) {
        int col = h * 64 + nt * 16 + n16;
#pragma unroll
        for (int r = 0; r < 8; ++r) {
            int row = i * 64 + w * 16 + r + 8 * half;
            ctx[(size_t)row * HIDN + col] = of[nt][r] / lrow[r];
        }
    }
}

// ---------------------------------------------------------------------------
extern "C" void kernel_launch(void* const* d_in, const int* in_sizes, int n_in,
                              void* d_out, int out_size, void* d_ws, size_t ws_size,
                              hipStream_t stream) {
    const float* hs      = (const float*)d_in[0];
    const float* w_qkv   = (const float*)d_in[1];
    const float* b_qkv   = (const float*)d_in[2];
    const float* w_dense = (const float*)d_in[3];
    const float* b_dense = (const float*)d_in[4];
    float* out = (float*)d_out;

    float* ws    = (float*)d_ws;
    float* mixed = ws;                                     // 2048*3072
    float* qr    = mixed + (size_t)S_LEN * NQKV;           // 32*2048*64
    float* kr    = qr    + (size_t)NHQ  * S_LEN * DH;      // 8*2048*64
    float* vvp   = kr    + (size_t)NHKV * S_LEN * DH;      // 8*2048*64
    float* ctx   = vvp   + (size_t)NHKV * S_LEN * DH;      // 2048*2048

    // 1) QKV projection
    gemm_nt_bias<<<dim3(NQKV / 64, S_LEN / 64), 128, 0, stream>>>(
        hs, w_qkv, b_qkv, mixed, S_LEN, NQKV, HIDN);
    // 2) RoPE + scatter
    rope_scatter<<<(S_LEN * NQKV + 255) / 256, 256, 0, stream>>>(mixed, qr, kr, vvp);
    // 3) block-sparse flash attention
    attn_kernel<<<dim3(NBLK, NHQ), 128, 0, stream>>>(qr, kr, vvp, ctx);
    // 4) dense output projection
    gemm_nt_bias<<<dim3(HIDN / 64, S_LEN / 64), 128, 0, stream>>>(
        ctx, w_dense, b_dense, out, S_LEN, HIDN, HIDN);
}